// Linear_86535001079936
// MI455X (gfx1250) — compile-verified
//
#include <hip/hip_runtime.h>
#include <hip/hip_bf16.h>

// CDNA5 (gfx1250) wave32 WMMA implementation of the grouped linear op.
// D = A(16x4) * B(4x16) + C via V_WMMA_F32_16X16X4_F32, fp32 end to end.

typedef __attribute__((ext_vector_type(2))) float v2f;
typedef __attribute__((ext_vector_type(8))) float v8f;

constexpr int kMUL0 = 128;
constexpr int kMUL1 = 64;
constexpr int kMUL2 = 32;
constexpr int kOFF1 = 128;           // MUL0*D0
constexpr int kOFF2 = 320;           // OFF1 + MUL1*3
constexpr int kDIM  = 480;           // OFF2 + MUL2*5
constexpr int LSTR  = 484;           // padded LDS row stride (dwords); 16B aligned,
                                     // 36*r mod 64 distinct for r=0..15 -> conflict-free column reads

__global__ void __launch_bounds__(32)
grouped_linear_wmma_f32(const float* __restrict__ x,
                        const float* __restrict__ w0,
                        const float* __restrict__ w1,
                        const float* __restrict__ w2,
                        const float* __restrict__ b0,
                        float* __restrict__ out,
                        int n_rows)
{
    __shared__ float X[16 * LSTR];

    const int lane = threadIdx.x & 31;
    const int lh   = lane >> 4;      // lane-half: selects K pair {0,1} vs {2,3}
    const int ln   = lane & 15;      // M (A) / N (B,C,D) index within half
    const long long row0 = (long long)blockIdx.x * 16;

    const float* gx = x   + row0 * kDIM;
    float*       gy = out + row0 * kDIM;

    // valid float4 chunks in this tile (16 rows * 120 float4 per row)
    int valid = (int)((long long)n_rows - row0) * (kDIM / 4);
    if (valid > 16 * (kDIM / 4)) valid = 16 * (kDIM / 4);

    // ---- coalesced copy-in: 16x480 contiguous floats -> padded LDS panel ----
    for (int t = 0; t < 60; ++t) {
        int e = t * 32 + lane;               // float4 index 0..1919
        if (e < valid) {
            float4 v = ((const float4*)gx)[e];
            int r = e / 120;
            int c = e % 120;
            *(float4*)&X[r * LSTR + c * 4] = v;
        }
    }
    __syncthreads();

    const float a0s = 0.088388347648318447f;  // 1/sqrt(128)
    const float a1s = 0.125f;                 // 1/sqrt(64)
    const float a2s = 0.176776695296636881f;  // 1/sqrt(32)

    // ================= Part 0: out0 = x0 @ (a0*w0) + b0, K=128, N=128 ============
    {
        v2f af[32];
        #pragma unroll
        for (int k = 0; k < 32; ++k) {
            int kk = 4 * k + 2 * lh;
            af[k].x = X[ln * LSTR + kk];
            af[k].y = X[ln * LSTR + kk + 1];
        }
        #pragma unroll 1
        for (int n0 = 0; n0 < 8; ++n0) {
            int n = n0 * 16 + ln;
            float bias = b0[n];
            v8f c;
            #pragma unroll
            for (int r = 0; r < 8; ++r) c[r] = bias;
            #pragma unroll
            for (int k = 0; k < 32; ++k) {
                int kk = 4 * k + 2 * lh;
                v2f b;
                b.x = a0s * w0[kk * kMUL0 + n];
                b.y = a0s * w0[(kk + 1) * kMUL0 + n];
                c = __builtin_amdgcn_wmma_f32_16x16x4_f32(
                        false, af[k], false, b, (short)0, c, false, false);
            }
            // A for part0 is register-resident: safe to overwrite cols 0..127
            #pragma unroll
            for (int r = 0; r < 8; ++r)
                X[(r + 8 * lh) * LSTR + n] = c[r];
        }
    }

    // ============ Part 1: 3 interleaved (16x64)x(64x64) GEMMs, stride-3 ==========
    #pragma unroll 1
    for (int i = 0; i < 3; ++i) {
        v2f af[16];
        #pragma unroll
        for (int k = 0; k < 16; ++k) {
            int kk = 4 * k + 2 * lh;
            af[k].x = X[ln * LSTR + kOFF1 + 3 * kk + i];
            af[k].y = X[ln * LSTR + kOFF1 + 3 * (kk + 1) + i];
        }
        #pragma unroll 1
        for (int n0 = 0; n0 < 4; ++n0) {
            int n = n0 * 16 + ln;
            v8f c = {};
            #pragma unroll
            for (int k = 0; k < 16; ++k) {
                int kk = 4 * k + 2 * lh;
                v2f b;
                b.x = a1s * w1[kk * kMUL1 + n];
                b.y = a1s * w1[(kk + 1) * kMUL1 + n];
                c = __builtin_amdgcn_wmma_f32_16x16x4_f32(
                        false, af[k], false, b, (short)0, c, false, false);
            }
            // write set {OFF1 + 3w + i} equals this i's read set, reads are done
            #pragma unroll
            for (int r = 0; r < 8; ++r)
                X[(r + 8 * lh) * LSTR + kOFF1 + 3 * n + i] = c[r];
        }
    }

    // ============ Part 2: 5 interleaved (16x32)x(32x32) GEMMs, stride-5 ==========
    #pragma unroll 1
    for (int i = 0; i < 5; ++i) {
        v2f af[8];
        #pragma unroll
        for (int k = 0; k < 8; ++k) {
            int kk = 4 * k + 2 * lh;
            af[k].x = X[ln * LSTR + kOFF2 + 5 * kk + i];
            af[k].y = X[ln * LSTR + kOFF2 + 5 * (kk + 1) + i];
        }
        #pragma unroll 1
        for (int n0 = 0; n0 < 2; ++n0) {
            int n = n0 * 16 + ln;
            v8f c = {};
            #pragma unroll
            for (int k = 0; k < 8; ++k) {
                int kk = 4 * k + 2 * lh;
                v2f b;
                b.x = a2s * w2[kk * kMUL2 + n];
                b.y = a2s * w2[(kk + 1) * kMUL2 + n];
                c = __builtin_amdgcn_wmma_f32_16x16x4_f32(
                        false, af[k], false, b, (short)0, c, false, false);
            }
            #pragma unroll
            for (int r = 0; r < 8; ++r)
                X[(r + 8 * lh) * LSTR + kOFF2 + 5 * n + i] = c[r];
        }
    }

    __syncthreads();

    // ---- coalesced copy-out: padded LDS panel -> 16x480 contiguous floats ----
    for (int t = 0; t < 60; ++t) {
        int e = t * 32 + lane;
        if (e < valid) {
            int r = e / 120;
            int c = e % 120;
            float4 v = *(const float4*)&X[r * LSTR + c * 4];
            ((float4*)gy)[e] = v;
        }
    }
}

extern "C" void kernel_launch(void* const* d_in, const int* in_sizes, int n_in,
                              void* d_out, int out_size, void* d_ws, size_t ws_size,
                              hipStream_t stream) {
    (void)n_in; (void)out_size; (void)d_ws; (void)ws_size;
    const float* x  = (const float*)d_in[0];
    const float* w0 = (const float*)d_in[1];
    const float* w1 = (const float*)d_in[2];
    const float* w2 = (const float*)d_in[3];
    const float* b0 = (const float*)d_in[4];
    float* out = (float*)d_out;

    int n_rows = in_sizes[0] / kDIM;          // 200000
    int blocks = (n_rows + 15) / 16;          // 12500, one 16-row tile per 1-wave block
    grouped_linear_wmma_f32<<<blocks, 32, 0, stream>>>(x, w0, w1, w2, b0, out, n_rows);
}